// DeformableSimulator_41154376630481
// MI455X (gfx1250) — compile-verified
//
#include <hip/hip_runtime.h>
#include <hip/hip_bf16.h>

typedef __attribute__((ext_vector_type(2))) float v2f;
typedef __attribute__((ext_vector_type(4))) float v4f;
typedef __attribute__((ext_vector_type(8))) float v8f;

// ---------------------------------------------------------------------------
// Kernel 1: padded delta4[v] = {dx, dy, dz, 0} where
//   d = next_position - (position + velocity*dt + accel*dt^2)
// ---------------------------------------------------------------------------
__global__ __launch_bounds__(256) void compute_delta4_kernel(
    const float* __restrict__ nextp, const float* __restrict__ pos,
    const float* __restrict__ vel, const float* __restrict__ acc,
    const float* __restrict__ dtp, float* __restrict__ delta4, int V) {
  int v = blockIdx.x * blockDim.x + threadIdx.x;
  if (v < V) {
    const float dt = dtp[0];
    v4f d;
#pragma unroll
    for (int c = 0; c < 3; ++c) {
      const int i = v * 3 + c;
      d[c] = nextp[i] - (pos[i] + vel[i] * dt + acc[i] * dt * dt);
    }
    d[3] = 0.0f;
    *(v4f*)(delta4 + (size_t)v * 4) = d;   // global_store_b128
  }
}

// ---------------------------------------------------------------------------
// Kernel 2: kinetic partials.
// kinetic_raw = sum_{i,j} M[i,j] * dot(delta_i, delta_j)
// Gram tile G (16x16) per column block via V_WMMA_F32_16X16X4_F32, then
// elementwise multiply with the streamed M tile (non-temporal; zero reuse).
// 4 independent accumulators break the fmac RAW chain.
// ---------------------------------------------------------------------------
template <int BPC>
__global__ __launch_bounds__(32) void kinetic_partial_tpl(
    const float* __restrict__ M, const float* __restrict__ delta4,
    float* __restrict__ partials, int V) {
  const int ib    = blockIdx.x;          // row block (16 rows)
  const int chunk = blockIdx.y;          // column chunk
  const int lane  = threadIdx.x;         // 0..31
  const int half  = lane >> 4;           // 0: K={0,1} slab, 1: K={2,pad}
  const int l     = lane & 15;

  // A tile: one uniform b64 load (lanes 0-15: {d0,d1}, lanes 16-31: {d2,0}).
  const v2f a = *(const v2f*)(delta4 + ((size_t)(ib * 16 + l) * 4 + half * 2));

  const int jb0 = chunk * BPC;

  // Loop-invariant bases: 8 M-row streams, prefetch stream, B-tile stream.
  const float* mbase = M + (size_t)(ib * 16 + half * 8) * (size_t)V
                         + (size_t)(jb0 * 16 + l);
  // Prefetch: lane -> row ib*16+(lane&15); half selects +1 / +2 tiles ahead,
  // so one prefetch_b8 covers the next cacheline of all 16 row streams.
  const float* pfbase = M + (size_t)(ib * 16 + l) * (size_t)V
                          + (size_t)((jb0 + 1 + half) * 16);
  const float* bbase = delta4 + ((size_t)(jb0 * 16 + l) * 4 + half * 2);

  float acc0 = 0.0f, acc1 = 0.0f, acc2 = 0.0f, acc3 = 0.0f;
  v2f b = *(const v2f*)bbase;            // B tile for t = 0

#pragma unroll
  for (int t = 0; t < BPC; ++t) {
    __builtin_prefetch(pfbase + t * 16, 0, 3);   // global_prefetch_b8, imm off

    float m[8];
#pragma unroll
    for (int v = 0; v < 8; ++v) {
      m[v] = __builtin_nontemporal_load(mbase + (size_t)v * (size_t)V + t * 16);
    }

    // G = A x B : G[mm][n] = dot(delta[ib*16+mm], delta[jb0*16+t*16+n])
    v8f c = {};
    v8f g = __builtin_amdgcn_wmma_f32_16x16x4_f32(
        /*neg_a=*/false, a, /*neg_b=*/false, b,
        /*c_mod=*/(short)0, c, /*reuse_a=*/false, /*reuse_b=*/false);

    if (t + 1 < BPC) {                   // compile-time guard after unroll
      b = *(const v2f*)(bbase + (size_t)(t + 1) * 64);   // imm offset b64
    }

    // 4 independent accumulator chains (no back-to-back RAW on one VGPR).
    acc0 = fmaf(m[0], g[0], acc0);
    acc1 = fmaf(m[1], g[1], acc1);
    acc2 = fmaf(m[2], g[2], acc2);
    acc3 = fmaf(m[3], g[3], acc3);
    acc0 = fmaf(m[4], g[4], acc0);
    acc1 = fmaf(m[5], g[5], acc1);
    acc2 = fmaf(m[6], g[6], acc2);
    acc3 = fmaf(m[7], g[7], acc3);
  }

  float acc = (acc0 + acc1) + (acc2 + acc3);

  // Deterministic wave32 reduction.
#pragma unroll
  for (int off = 16; off > 0; off >>= 1) acc += __shfl_down(acc, off, 32);
  if (lane == 0) partials[(size_t)chunk * gridDim.x + ib] = acc;
}

// Generic fallback (runtime trip count) for unexpected V.
__global__ __launch_bounds__(32) void kinetic_partial_generic(
    const float* __restrict__ M, const float* __restrict__ delta4,
    float* __restrict__ partials, int V, int blocksPerChunk) {
  const int ib    = blockIdx.x;
  const int chunk = blockIdx.y;
  const int lane  = threadIdx.x;
  const int half  = lane >> 4;
  const int l     = lane & 15;

  const v2f a = *(const v2f*)(delta4 + ((size_t)(ib * 16 + l) * 4 + half * 2));
  const int jb0 = chunk * blocksPerChunk;
  float acc = 0.0f;
  for (int t = 0; t < blocksPerChunk; ++t) {
    const int jb = jb0 + t;
    const v2f b = *(const v2f*)(delta4 + ((size_t)(jb * 16 + l) * 4 + half * 2));
    const float* Mrow = M + (size_t)(ib * 16 + half * 8) * (size_t)V
                          + (size_t)(jb * 16 + l);
    float m[8];
#pragma unroll
    for (int v = 0; v < 8; ++v)
      m[v] = __builtin_nontemporal_load(Mrow + (size_t)v * (size_t)V);
    v8f c = {};
    v8f g = __builtin_amdgcn_wmma_f32_16x16x4_f32(
        false, a, false, b, (short)0, c, false, false);
#pragma unroll
    for (int v = 0; v < 8; ++v) acc = fmaf(m[v], g[v], acc);
  }
#pragma unroll
  for (int off = 16; off > 0; off >>= 1) acc += __shfl_down(acc, off, 32);
  if (lane == 0) partials[(size_t)chunk * gridDim.x + ib] = acc;
}

// ---------------------------------------------------------------------------
// Kernel 3: elastic partials (one thread per tetrahedral element).
// ---------------------------------------------------------------------------
__global__ __launch_bounds__(256) void elastic_partial_kernel(
    const float* __restrict__ nextp, const long long* __restrict__ elems,
    const float* __restrict__ poly, const float* __restrict__ meas,
    const float* __restrict__ lam, const float* __restrict__ mu,
    float* __restrict__ partials, int E) {
  __shared__ float sdata[256];
  const int e = blockIdx.x * blockDim.x + threadIdx.x;
  float val = 0.0f;
  if (e < E) {
    float F[3][3] = {{0.f,0.f,0.f},{0.f,0.f,0.f},{0.f,0.f,0.f}};
#pragma unroll
    for (int f = 0; f < 4; ++f) {
      const long long vi = elems[(size_t)e * 4 + f];
      const float px = nextp[vi * 3 + 0];
      const float py = nextp[vi * 3 + 1];
      const float pz = nextp[vi * 3 + 2];
#pragma unroll
      for (int l = 0; l < 3; ++l) {
        const float p = poly[(size_t)e * 16 + f * 4 + l];
        F[0][l] = fmaf(px, p, F[0][l]);
        F[1][l] = fmaf(py, p, F[1][l]);
        F[2][l] = fmaf(pz, p, F[2][l]);
      }
    }
    float Ic = 0.0f;
#pragma unroll
    for (int i = 0; i < 3; ++i)
#pragma unroll
      for (int j = 0; j < 3; ++j) Ic = fmaf(F[i][j], F[i][j], Ic);

    const float J =
        F[0][0] * (F[1][1] * F[2][2] - F[1][2] * F[2][1]) -
        F[0][1] * (F[1][0] * F[2][2] - F[1][2] * F[2][0]) +
        F[0][2] * (F[1][0] * F[2][1] - F[1][1] * F[2][0]);

    const float lm = lam[e];
    const float m  = mu[e];
    const float alpha = 0.75f * m / lm + 1.0f;   // (1 - 1/4)*mu/lam + 1
    float icv = Ic + 1.0f;                       // relu(Ic + delta)
    icv = icv > 0.0f ? icv : 0.0f;
    const float d = J - alpha;
    const float psi = 0.5f * m * (Ic - 3.0f) + 0.5f * lm * d * d
                    - 0.5f * m * __logf(icv + 1e-30f);
    val = psi * meas[(size_t)e * 4 + 3];
  }
  sdata[threadIdx.x] = val;
  __syncthreads();
#pragma unroll
  for (int s = 128; s > 0; s >>= 1) {
    if (threadIdx.x < (unsigned)s) sdata[threadIdx.x] += sdata[threadIdx.x + s];
    __syncthreads();
  }
  if (threadIdx.x == 0) partials[blockIdx.x] = sdata[0];
}

// ---------------------------------------------------------------------------
// Kernel 4: deterministic finalize -> (energy, kinetic, elastic)
// ---------------------------------------------------------------------------
__global__ __launch_bounds__(256) void finalize_kernel(
    const float* __restrict__ kin_partials, int nk,
    const float* __restrict__ el_partials, int ne,
    const float* __restrict__ dtp, float* __restrict__ out) {
  __shared__ float sk[256];
  __shared__ float se[256];
  float k = 0.0f, el = 0.0f;
  for (int i = threadIdx.x; i < nk; i += 256) k  += kin_partials[i];
  for (int i = threadIdx.x; i < ne; i += 256) el += el_partials[i];
  sk[threadIdx.x] = k;
  se[threadIdx.x] = el;
  __syncthreads();
#pragma unroll
  for (int s = 128; s > 0; s >>= 1) {
    if (threadIdx.x < (unsigned)s) {
      sk[threadIdx.x] += sk[threadIdx.x + s];
      se[threadIdx.x] += se[threadIdx.x + s];
    }
    __syncthreads();
  }
  if (threadIdx.x == 0) {
    const float dt = dtp[0];
    const float inv_h = 1.0f / dt;
    const float coeff = 0.5f * inv_h * inv_h;
    const float kin = coeff * sk[0];
    const float ela = se[0];
    out[0] = kin + ela;   // energy
    out[1] = kin;         // kinetic
    out[2] = ela;         // elastic
  }
}

// ---------------------------------------------------------------------------
// Launch
// ---------------------------------------------------------------------------
extern "C" void kernel_launch(void* const* d_in, const int* in_sizes, int n_in,
                              void* d_out, int out_size, void* d_ws, size_t ws_size,
                              hipStream_t stream) {
  const float*     nextp = (const float*)d_in[0];
  const float*     pos   = (const float*)d_in[1];
  const float*     vel   = (const float*)d_in[2];
  const float*     acc   = (const float*)d_in[3];
  const float*     M     = (const float*)d_in[4];
  const long long* elems = (const long long*)d_in[5];
  const float*     poly  = (const float*)d_in[6];
  const float*     meas  = (const float*)d_in[7];
  const float*     lam   = (const float*)d_in[8];
  const float*     mu    = (const float*)d_in[9];
  const float*     dtp   = (const float*)d_in[10];
  float* out = (float*)d_out;

  const int V = in_sizes[0] / 3;           // 8192
  const int E = in_sizes[5] / 4;           // 40000

  // Workspace layout (floats): [delta4 V*4][kin partials][elastic partials]
  float* delta4 = (float*)d_ws;
  const int nRowBlocks = V / 16;           // 512
  const int nChunks    = 32;
  const int blocksPerChunk = nRowBlocks / nChunks;  // 16 for V=8192
  const int nk = nRowBlocks * nChunks;     // 16384 kinetic partials
  float* kin_partials = delta4 + (size_t)V * 4;
  const int eBlocks = (E + 255) / 256;     // elastic partial count
  float* el_partials = kin_partials + nk;

  // 1) padded delta
  compute_delta4_kernel<<<(V + 255) / 256, 256, 0, stream>>>(
      nextp, pos, vel, acc, dtp, delta4, V);

  // 2) kinetic partials (WMMA Gram tiles x streamed M tiles)
  {
    dim3 grid(nRowBlocks, nChunks);
    if (blocksPerChunk == 16) {
      kinetic_partial_tpl<16><<<grid, 32, 0, stream>>>(M, delta4, kin_partials, V);
    } else if (blocksPerChunk == 8) {
      kinetic_partial_tpl<8><<<grid, 32, 0, stream>>>(M, delta4, kin_partials, V);
    } else {
      kinetic_partial_generic<<<grid, 32, 0, stream>>>(
          M, delta4, kin_partials, V, blocksPerChunk);
    }
  }
  // 3) elastic partials
  elastic_partial_kernel<<<eBlocks, 256, 0, stream>>>(
      nextp, elems, poly, meas, lam, mu, el_partials, E);

  // 4) finalize
  finalize_kernel<<<1, 256, 0, stream>>>(kin_partials, nk, el_partials, eBlocks,
                                         dtp, out);
}